// FasterRCNN_17927193493949
// MI455X (gfx1250) — compile-verified
//
#include <hip/hip_runtime.h>
#include <hip/hip_bf16.h>
#include <stdint.h>

#define N_ROI   2000
#define N_PAD   2048          // padded, power of two for bitonic sort
#define N_CLASS 20
#define N_C1    21
#define IMG_HW  800.0f
#define SCORE_TH 0.05f
#define NMS_TH   0.5f
#define NWORD16  128          // 2048 / 16 halfwords per mask row
#define NWORD32  64           // 2048 / 32 words per mask row

typedef __attribute__((ext_vector_type(2))) float v2f;
typedef __attribute__((ext_vector_type(8))) float v8f;

// ---------------------------------------------------------------------------
// Kernel 1: softmax over 21 classes + box decode for classes 1..20.
// dec  : [20][2000][4]  (class-major decoded boxes)
// prob : [20][2000]
// ---------------------------------------------------------------------------
__global__ __launch_bounds__(128) void k1_decode(
    const float* __restrict__ score, const float* __restrict__ loc,
    const float* __restrict__ rois,  const float* __restrict__ mean,
    const float* __restrict__ stdv,
    float* __restrict__ dec, float* __restrict__ prob)
{
    int r = blockIdx.x * blockDim.x + threadIdx.x;
    if (r >= N_ROI) return;

    float sc[N_C1];
    float mx = -__builtin_inff();
    #pragma unroll
    for (int c = 0; c < N_C1; ++c) { sc[c] = score[r * N_C1 + c]; mx = fmaxf(mx, sc[c]); }
    float sum = 0.f;
    #pragma unroll
    for (int c = 0; c < N_C1; ++c) { sc[c] = __expf(sc[c] - mx); sum += sc[c]; }
    float inv = 1.f / sum;

    float y1 = rois[r * 4 + 0], x1 = rois[r * 4 + 1];
    float y2 = rois[r * 4 + 2], x2 = rois[r * 4 + 3];
    float h = y2 - y1, w = x2 - x1;
    float cy = y1 + 0.5f * h, cx = x1 + 0.5f * w;
    float m0 = mean[0], m1 = mean[1], m2 = mean[2], m3 = mean[3];
    float s0 = stdv[0], s1 = stdv[1], s2 = stdv[2], s3 = stdv[3];

    for (int c = 1; c < N_C1; ++c) {
        const float* lp = loc + (size_t)r * (N_C1 * 4) + c * 4;
        float dy = lp[0] * s0 + m0, dx = lp[1] * s1 + m1;
        float dh = lp[2] * s2 + m2, dw = lp[3] * s3 + m3;
        float ncy = dy * h + cy, ncx = dx * w + cx;
        float nh = __expf(dh) * h, nw = __expf(dw) * w;
        float ymin = fminf(fmaxf(ncy - 0.5f * nh, 0.f), IMG_HW);
        float xmin = fminf(fmaxf(ncx - 0.5f * nw, 0.f), IMG_HW);
        float ymax = fminf(fmaxf(ncy + 0.5f * nh, 0.f), IMG_HW);
        float xmax = fminf(fmaxf(ncx + 0.5f * nw, 0.f), IMG_HW);
        float* dp = dec + ((size_t)(c - 1) * N_ROI + r) * 4;
        dp[0] = ymin; dp[1] = xmin; dp[2] = ymax; dp[3] = xmax;
        prob[(size_t)(c - 1) * N_ROI + r] = sc[c] * inv;
    }
}

// ---------------------------------------------------------------------------
// Kernel 2: per-class descending bitonic sort of (key,idx) over 2048 slots.
// key = valid ? prob : -inf  (reference: argsort(where(valid,s,-inf))[::-1]).
// Emits sorted boxes, scores, valid flags, and areas.
// ---------------------------------------------------------------------------
__global__ __launch_bounds__(1024) void k2_sort(
    const float* __restrict__ prob, const float* __restrict__ dec,
    float4* __restrict__ sortedBox, float* __restrict__ sortedScore,
    unsigned* __restrict__ sortedValid, float* __restrict__ areaArr)
{
    __shared__ float key[N_PAD];
    __shared__ int   idxs[N_PAD];
    const int c = blockIdx.x;
    const int tid = threadIdx.x;

    for (int e = tid; e < N_PAD; e += 1024) {
        float k = -__builtin_inff();
        if (e < N_ROI) {
            float p = prob[(size_t)c * N_ROI + e];
            if (p > SCORE_TH) k = p;
        }
        key[e] = k; idxs[e] = e;
    }
    __syncthreads();

    for (int k = 2; k <= N_PAD; k <<= 1) {
        for (int j = k >> 1; j > 0; j >>= 1) {
            for (int t = tid; t < N_PAD; t += 1024) {
                int ixj = t ^ j;
                if (ixj > t) {
                    bool dirDesc = ((t & k) == 0);
                    float a0 = key[t], a1 = key[ixj];
                    bool doSwap = dirDesc ? (a0 < a1) : (a0 > a1);
                    if (doSwap) {
                        key[t] = a1; key[ixj] = a0;
                        int i0 = idxs[t]; idxs[t] = idxs[ixj]; idxs[ixj] = i0;
                    }
                }
            }
            __syncthreads();
        }
    }

    for (int e = tid; e < N_PAD; e += 1024) {
        int si = idxs[e];
        float kkey = key[e];
        bool validE = kkey > -__builtin_inff();
        float4 bx; bx.x = 0.f; bx.y = 0.f; bx.z = 0.f; bx.w = 0.f;
        float p = 0.f;
        if (si < N_ROI) {
            const float* dp = dec + ((size_t)c * N_ROI + si) * 4;
            bx.x = dp[0]; bx.y = dp[1]; bx.z = dp[2]; bx.w = dp[3];
            p = prob[(size_t)c * N_ROI + si];
        }
        size_t o = (size_t)c * N_PAD + e;
        sortedBox[o]   = bx;
        sortedScore[o] = p;
        sortedValid[o] = validE ? 1u : 0u;
        areaArr[o]     = (bx.z - bx.x) * (bx.w - bx.y);
    }
}

// ---------------------------------------------------------------------------
// Kernel 3: parallel suppression-mask build.
// Each wave32 owns a 16-row i-block; loops over 128 j-chunks of 16.
// area_i + area_j for the 16x16 tile computed with v_wmma_f32_16x16x4_f32
// (A = [area_i, 1] 16x4, B = [1 ; area_j] 4x16) — D-layout lane=N, vgpr(+8*half)=M
// matches the per-lane VALU intersection terms. Bits packed via ballot_w32.
// Class boxes/areas staged into LDS via global_load_async_to_lds_b128.
// mask: u16 [20][2048][128], bit b of halfword jc of row i => j = jc*16+b.
// ---------------------------------------------------------------------------
__global__ __launch_bounds__(256) void k3_mask(
    const float4* __restrict__ sortedBox, const float* __restrict__ areaArr,
    unsigned short* __restrict__ mask)
{
    __shared__ float4 box_s[N_PAD];     // 32 KB
    __shared__ float  area_s[N_PAD];    // 8 KB
    const int c    = blockIdx.y;
    const int tid  = threadIdx.x;

    // ---- async stage: 2048 float4 boxes + 2048 float areas -> LDS ----
    const float4* gbox = sortedBox + (size_t)c * N_PAD;
    #pragma unroll
    for (int rr = 0; rr < 8; ++rr) {
        int e = rr * 256 + tid;
        unsigned loff = (unsigned)(uintptr_t)(&box_s[e]);
        const float4* g = gbox + e;
        asm volatile("global_load_async_to_lds_b128 %0, %1, off"
                     :: "v"(loff), "v"(g) : "memory");
    }
    const float4* garea4 = (const float4*)(areaArr + (size_t)c * N_PAD);
    #pragma unroll
    for (int rr = 0; rr < 2; ++rr) {
        int e = rr * 256 + tid;
        unsigned loff = (unsigned)(uintptr_t)(&area_s[e * 4]);
        const float4* g = garea4 + e;
        asm volatile("global_load_async_to_lds_b128 %0, %1, off"
                     :: "v"(loff), "v"(g) : "memory");
    }
    asm volatile("s_wait_asynccnt 0" ::: "memory");
    __syncthreads();

    const int wv     = tid >> 5;
    const int lane   = tid & 31;
    const int laneN  = lane & 15;
    const bool loHalf = lane < 16;
    const int iBase  = (blockIdx.x * 8 + wv) * 16;
    const int iLane  = iBase + (loHalf ? 0 : 8);   // M row base this half-lane sees

    // A operand: 16x4, A[M][0]=area_i[M], A[M][1]=1 (lanes 0-15: K=0,1; 16-31: K=2,3 = 0)
    v2f a;
    a.x = loHalf ? area_s[iBase + lane] : 0.0f;
    a.y = loHalf ? 1.0f : 0.0f;

    float4 bi[8];
    #pragma unroll
    for (int v = 0; v < 8; ++v) bi[v] = box_s[iLane + v];

    unsigned short* mrow = mask + (size_t)c * N_PAD * NWORD16;

    for (int jc = 0; jc < NWORD16; ++jc) {
        const int jBase = jc * 16;
        const int jpos  = jBase + laneN;
        float4 bj = box_s[jBase + laneN];
        float  aj = loHalf ? area_s[jBase + lane] : 0.0f;

        // B operand: 4x16, B[0][N]=1, B[1][N]=area_j[N]
        v2f b; b.x = loHalf ? 1.0f : 0.0f; b.y = aj;
        v8f cz = {0.f, 0.f, 0.f, 0.f, 0.f, 0.f, 0.f, 0.f};
        v8f asum = __builtin_amdgcn_wmma_f32_16x16x4_f32(
            false, a, false, b, (short)0, cz, false, false);

        unsigned bal[8];
        #pragma unroll
        for (int v = 0; v < 8; ++v) {
            float4 bb = bi[v];
            float ih = fminf(bb.z, bj.z) - fmaxf(bb.x, bj.x);
            float iw = fminf(bb.w, bj.w) - fmaxf(bb.y, bj.y);
            ih = fmaxf(ih, 0.f); iw = fmaxf(iw, 0.f);
            float inter = ih * iw;
            int ipos = iLane + v;
            // iou > th  <=>  inter > th * (area_i+area_j - inter + eps)
            bool pred = (inter > NMS_TH * (asum[v] - inter + 1e-9f)) && (jpos > ipos);
            bal[v] = __builtin_amdgcn_ballot_w32(pred);
        }
        if (lane == 0) {
            #pragma unroll
            for (int v = 0; v < 8; ++v) {
                mrow[(size_t)(iBase + v)     * NWORD16 + jc] = (unsigned short)(bal[v] & 0xFFFFu);
                mrow[(size_t)(iBase + v + 8) * NWORD16 + jc] = (unsigned short)(bal[v] >> 16);
            }
        }
    }
}

// ---------------------------------------------------------------------------
// Kernel 4: serial suppression reduce — one wave32 per class.
// Running 2048-bit suppressed mask lives in LDS (64 u32, 2 per lane).
// keep[i] = valid[i] && !suppressed[i]; survivors OR their mask row in.
// ---------------------------------------------------------------------------
__global__ __launch_bounds__(32) void k4_reduce(
    const unsigned short* __restrict__ mask,
    const unsigned* __restrict__ sortedValid, unsigned* __restrict__ keepFlag)
{
    __shared__ unsigned sup[NWORD32];
    const int c = blockIdx.x;
    const int lane = threadIdx.x;
    sup[lane] = 0u; sup[lane + 32] = 0u;
    __syncthreads();

    const unsigned* vflags = sortedValid + (size_t)c * N_PAD;
    unsigned* kf = keepFlag + (size_t)c * N_PAD;
    const unsigned* mbase = (const unsigned*)(mask + (size_t)c * N_PAD * NWORD16);

    for (int i = 0; i < N_PAD; ++i) {
        unsigned supw = sup[i >> 5];
        bool alive = (vflags[i] != 0u) && !((supw >> (i & 31)) & 1u);
        if (lane == 0) kf[i] = alive ? 1u : 0u;
        if (alive) {
            const unsigned* row = mbase + (size_t)i * NWORD32;
            sup[lane]      |= row[lane];
            sup[lane + 32] |= row[lane + 32];
        }
        __syncthreads();
    }
}

// ---------------------------------------------------------------------------
// Kernel 5: scatter outputs (in sorted order, matching reference return).
// out = [bboxes 20*2000*4][scores 20*2000][labels 20*2000][keep 20*2000]
// ---------------------------------------------------------------------------
__global__ __launch_bounds__(256) void k5_out(
    const float4* __restrict__ sortedBox, const float* __restrict__ sortedScore,
    const unsigned* __restrict__ keepFlag, float* __restrict__ out)
{
    int idx = blockIdx.x * blockDim.x + threadIdx.x;
    if (idx >= N_CLASS * N_ROI) return;
    int c = idx / N_ROI, r = idx - c * N_ROI;
    size_t e = (size_t)c * N_PAD + r;
    unsigned k = keepFlag[e];
    float4 b = sortedBox[e];
    float  s = sortedScore[e];
    if (!k) { b.x = 0.f; b.y = 0.f; b.z = 0.f; b.w = 0.f; s = 0.f; }
    float* ob = out + (size_t)idx * 4;
    ob[0] = b.x; ob[1] = b.y; ob[2] = b.z; ob[3] = b.w;
    out[160000 + idx] = s;
    out[200000 + idx] = (float)(c + 1);
    out[240000 + idx] = k ? 1.f : 0.f;
}

// ---------------------------------------------------------------------------
extern "C" void kernel_launch(void* const* d_in, const int* in_sizes, int n_in,
                              void* d_out, int out_size, void* d_ws, size_t ws_size,
                              hipStream_t stream)
{
    const float* score = (const float*)d_in[0];
    const float* loc   = (const float*)d_in[1];
    const float* rois  = (const float*)d_in[2];
    const float* mean  = (const float*)d_in[3];
    const float* stdv  = (const float*)d_in[4];

    char* ws = (char*)d_ws;
    float*          dec         = (float*)         (ws + 0);          // 640000 B
    float*          prob        = (float*)         (ws + 640000);     // 160000 B
    float4*         sortedBox   = (float4*)        (ws + 800000);     // 655360 B
    float*          sortedScore = (float*)         (ws + 1455360);    // 163840 B
    unsigned*       sortedValid = (unsigned*)      (ws + 1619200);    // 163840 B
    float*          areaArr     = (float*)         (ws + 1783040);    // 163840 B
    unsigned*       keepFlag    = (unsigned*)      (ws + 1946880);    // 163840 B
    unsigned short* mask        = (unsigned short*)(ws + 2110720);    // 10485760 B

    float* out = (float*)d_out;

    k1_decode<<<dim3((N_ROI + 127) / 128), dim3(128), 0, stream>>>(
        score, loc, rois, mean, stdv, dec, prob);
    k2_sort<<<dim3(N_CLASS), dim3(1024), 0, stream>>>(
        prob, dec, sortedBox, sortedScore, sortedValid, areaArr);
    k3_mask<<<dim3(16, N_CLASS), dim3(256), 0, stream>>>(
        sortedBox, areaArr, mask);
    k4_reduce<<<dim3(N_CLASS), dim3(32), 0, stream>>>(
        mask, sortedValid, keepFlag);
    k5_out<<<dim3((N_CLASS * N_ROI + 255) / 256), dim3(256), 0, stream>>>(
        sortedBox, sortedScore, keepFlag, out);
}